// VectorQuantizer_85358180041006
// MI455X (gfx1250) — compile-verified
//
#include <hip/hip_runtime.h>
#include <hip/hip_bf16.h>

typedef float v2f __attribute__((ext_vector_type(2)));
typedef float v8f __attribute__((ext_vector_type(8)));

#define VQ_D      64
#define VQ_K      1024
#define VQ_HW     1024        // H*W = 32*32
#define VQ_NTOK   65536       // B*H*W = 64*1024
#define VQ_NELEM  4194304     // B*D*H*W
#define CB_STRIDE 68          // padded LDS row stride (floats): 68 mod 64 = 4 -> no bank conflicts

// ---------------------------------------------------------------------------
// Kernel 1: argmin over codebook using f32 WMMA (16x16x4, K accumulated x16)
// Block = 256 threads (8 waves). Each wave handles 16 tokens vs all 1024 codes.
// Dynamic LDS: padded codebook (1024*68 floats) + c2 (1024 floats) = 282,624 B.
// ---------------------------------------------------------------------------
__global__ __launch_bounds__(256) void vq_argmin_kernel(
    const float* __restrict__ z_e,      // [64,64,32,32] NCHW
    const float* __restrict__ cb,       // [1024,64]
    int*   __restrict__ idx_out,        // [65536] workspace (int)
    float* __restrict__ idx_f_out)      // [65536] final output (float-valued)
{
  extern __shared__ float smem[];
  float* s_cb = smem;                    // 1024 rows * CB_STRIDE floats
  float* s_c2 = smem + VQ_K * CB_STRIDE; // 1024 floats

  const int tid = threadIdx.x;

  // ---- Stage codebook into LDS with padded rows (float4 granularity) ----
  const float4* cb4 = reinterpret_cast<const float4*>(cb);
  #pragma unroll
  for (int j = 0; j < 64; ++j) {
    int i   = tid + j * 256;            // 16384 float4 total
    int row = i >> 4;                   // code index
    int q   = i & 15;                   // which float4 in row
    *reinterpret_cast<float4*>(s_cb + row * CB_STRIDE + q * 4) = cb4[i];
  }
  __syncthreads();

  // ---- Precompute ||c||^2 per code ----
  #pragma unroll
  for (int c = 0; c < 4; ++c) {
    int code = tid + c * 256;
    const float* row = s_cb + code * CB_STRIDE;
    float acc = 0.f;
    #pragma unroll
    for (int j = 0; j < VQ_D; ++j) acc += row[j] * row[j];
    s_c2[code] = acc;
  }
  __syncthreads();

  const int wave   = tid >> 5;
  const int lane   = tid & 31;
  const int lane16 = lane & 15;
  const int hi     = lane >> 4;         // 0: K pair {0,1}; 1: K pair {2,3} (A layout)

  const int token_base = blockIdx.x * 128 + wave * 16;
  const int tok = token_base + lane16;  // row M = lane16 of the A tile
  const int b   = tok >> 10;
  const int s   = tok & 1023;
  const float* zb = z_e + b * (VQ_D * VQ_HW) + s;   // stride VQ_HW per feature dim

  // ---- Preload A: 16 tokens x 64 dims -> 16 x float2 per lane ----
  v2f a[16];
  #pragma unroll
  for (int kt = 0; kt < 16; ++kt) {
    int kofs = 4 * kt + 2 * hi;
    a[kt].x = zb[(kofs    ) * VQ_HW];
    a[kt].y = zb[(kofs + 1) * VQ_HW];
  }

  float best_val[8];
  int   best_idx[8];
  #pragma unroll
  for (int r = 0; r < 8; ++r) { best_val[r] = 3.4e38f; best_idx[r] = 0; }

  // ---- Sweep 64 code tiles of 16 codes each ----
  for (int t = 0; t < 64; ++t) {
    const int ncol = t * 16 + lane16;   // column N = lane16 of B/C tiles
    const float* brow = s_cb + ncol * CB_STRIDE + 2 * hi;
    v8f acc = {};
    #pragma unroll
    for (int kt = 0; kt < 16; ++kt) {
      // B layout (4x16 f32): lane holds column N=lane16, K pair per half-wave
      v2f bv = *reinterpret_cast<const v2f*>(brow + 4 * kt);  // ds_load_b64
      acc = __builtin_amdgcn_wmma_f32_16x16x4_f32(
          /*neg_a=*/false, a[kt], /*neg_b=*/false, bv,
          /*c_mod=*/(short)0, acc, /*reuse_a=*/false, /*reuse_b=*/false);
    }
    const float c2 = s_c2[ncol];
    #pragma unroll
    for (int r = 0; r < 8; ++r) {
      // d2' = ||c||^2 - 2 z.c  (||z||^2 constant per token -> argmin-equivalent)
      float d2 = c2 - 2.0f * acc[r];
      if (d2 < best_val[r]) { best_val[r] = d2; best_idx[r] = ncol; }
    }
  }

  // ---- Argmin reduction across the 16 lanes holding each row ----
  // xor masks 1,2,4,8 stay within each 16-lane group (rows 0-7 vs 8-15).
  #pragma unroll
  for (int m = 8; m >= 1; m >>= 1) {
    #pragma unroll
    for (int r = 0; r < 8; ++r) {
      float ov = __shfl_xor(best_val[r], m, 32);
      int   oi = __shfl_xor(best_idx[r], m, 32);
      if (ov < best_val[r] || (ov == best_val[r] && oi < best_idx[r])) {
        best_val[r] = ov; best_idx[r] = oi;
      }
    }
  }

  if (lane16 == 0) {
    #pragma unroll
    for (int r = 0; r < 8; ++r) {
      int row   = hi * 8 + r;           // C/D layout: vgpr r holds M=r (lo) / M=r+8 (hi)
      int token = token_base + row;
      idx_out[token]   = best_idx[r];
      idx_f_out[token] = (float)best_idx[r];
    }
  }
}

// ---------------------------------------------------------------------------
// Kernel 2: gather z_q = codebook[idx], write z_q_st (== z_q forward), and
// produce deterministic per-block sums of (z_q - z_e)^2.
// ---------------------------------------------------------------------------
__global__ __launch_bounds__(256) void vq_gather_kernel(
    const float* __restrict__ z_e,
    const float* __restrict__ cb,
    const int*   __restrict__ idx,
    float* __restrict__ out,            // z_q_st, NCHW flat
    float* __restrict__ block_sums)
{
  __shared__ float red[256];
  const int flat = blockIdx.x * 256 + threadIdx.x;   // < 4194304, = b*65536 + d*1024 + s
  const int s = flat & 1023;
  const int d = (flat >> 10) & 63;
  const int b = flat >> 16;
  const int n = (b << 10) | s;                       // token index

  const int   code = idx[n];
  const float zq   = cb[code * VQ_D + d];
  out[flat] = zq;                                    // coalesced store

  const float diff = zq - z_e[flat];                 // coalesced load
  red[threadIdx.x] = diff * diff;
  __syncthreads();
  #pragma unroll
  for (int st = 128; st >= 1; st >>= 1) {
    if (threadIdx.x < st) red[threadIdx.x] += red[threadIdx.x + st];
    __syncthreads();
  }
  if (threadIdx.x == 0) block_sums[blockIdx.x] = red[0];
}

// ---------------------------------------------------------------------------
// Kernel 3: deterministic final reduction of 16384 block sums -> loss scalar.
// loss = codebook_loss + 0.25*commit_loss = 1.25 * mean((z_q - z_e_perm)^2)
// ---------------------------------------------------------------------------
__global__ __launch_bounds__(256) void vq_loss_kernel(
    const float* __restrict__ block_sums,
    float* __restrict__ loss_out)
{
  __shared__ float red[256];
  float acc = 0.f;
  #pragma unroll
  for (int i = 0; i < 64; ++i) acc += block_sums[threadIdx.x + i * 256];
  red[threadIdx.x] = acc;
  __syncthreads();
  #pragma unroll
  for (int st = 128; st >= 1; st >>= 1) {
    if (threadIdx.x < st) red[threadIdx.x] += red[threadIdx.x + st];
    __syncthreads();
  }
  if (threadIdx.x == 0)
    loss_out[0] = 1.25f * red[0] / (float)VQ_NELEM;
}

extern "C" void kernel_launch(void* const* d_in, const int* in_sizes, int n_in,
                              void* d_out, int out_size, void* d_ws, size_t ws_size,
                              hipStream_t stream) {
  (void)in_sizes; (void)n_in; (void)out_size; (void)ws_size;

  const float* z_e = (const float*)d_in[0];   // 4,194,304 f32
  const float* cb  = (const float*)d_in[1];   // 65,536 f32

  float* out   = (float*)d_out;
  float* zq    = out;                         // [0, 4194304)
  float* loss  = out + VQ_NELEM;              // [4194304]
  float* idx_f = out + VQ_NELEM + 1;          // [4194305, +65536)

  int*   ws_idx   = (int*)d_ws;                                 // 65536 ints
  float* ws_bsums = (float*)((char*)d_ws + VQ_NTOK * sizeof(int)); // 16384 floats

  // Kernel 1: 512 blocks x 256 threads, 128 tokens/block.
  // Dynamic LDS: padded codebook + c2.
  const size_t smem_bytes = (size_t)(VQ_K * CB_STRIDE + VQ_K) * sizeof(float);
  vq_argmin_kernel<<<VQ_NTOK / 128, 256, smem_bytes, stream>>>(
      z_e, cb, ws_idx, idx_f);

  // Kernel 2: one thread per output element.
  vq_gather_kernel<<<VQ_NELEM / 256, 256, 0, stream>>>(
      z_e, cb, ws_idx, zq, ws_bsums);

  // Kernel 3: single-block deterministic reduction.
  vq_loss_kernel<<<1, 256, 0, stream>>>(ws_bsums, loss);
}